// nDropout_17540646437457
// MI455X (gfx1250) — compile-verified
//
#include <hip/hip_runtime.h>
#include <hip/hip_bf16.h>

typedef float v4f __attribute__((ext_vector_type(4)));
typedef int   v4i __attribute__((vector_size(16)));

// Problem constants from the reference: D = 2048 floats per row -> 512 float4.
#define DVC 512               // D / 4 (float4 chunks per row)
#define BLOCK_THREADS 512     // one float4 column chunk per thread, 16 wave32s

__global__ __launch_bounds__(BLOCK_THREADS) void dropout_mask_mul_kernel(
    const float* __restrict__ x,
    const float* __restrict__ mask,
    float* __restrict__ out,
    long rows,               // B*T
    int  rows_per_block)
{
    __shared__ v4f smask[DVC];

    const int t = (int)threadIdx.x;
    const v4f* __restrict__ xv = (const v4f*)x;
    const v4f* __restrict__ mv = (const v4f*)mask;
    v4f* __restrict__ ov       = (v4f*)out;

    // ---- Stage the 8 KB mask into LDS once per block ----
    // gfx1250 async global->LDS path (ASYNCcnt) when the toolchain exposes it;
    // plain cooperative copy otherwise.
#if defined(__gfx1250__) && __has_builtin(__builtin_amdgcn_global_load_async_to_lds_b128)
    {
        typedef __attribute__((address_space(1))) v4i gv4i_t;   // global (AS1)
        typedef __attribute__((address_space(3))) v4i lv4i_t;   // LDS (AS3)
        if (t < DVC) {
            __builtin_amdgcn_global_load_async_to_lds_b128(
                (gv4i_t*)(mv + t), (lv4i_t*)(smask + t), 0, 0);
        }
#if __has_builtin(__builtin_amdgcn_s_wait_asynccnt)
        __builtin_amdgcn_s_wait_asynccnt(0);
#else
        asm volatile("s_wait_asynccnt 0" ::: "memory");
#endif
    }
#else
    for (int c = t; c < DVC; c += (int)blockDim.x) {
        smask[c] = mv[c];
    }
#endif
    __syncthreads();

    // Thread t permanently owns float4 column chunk t: mask lives in registers.
    const v4f m = smask[t];

    // ---- Contiguous row range for this block ----
    long r0 = (long)blockIdx.x * (long)rows_per_block;
    long r1 = r0 + (long)rows_per_block;
    if (r1 > rows) r1 = rows;
    if (r0 >= r1) return;

    long i = r0 * (long)DVC + (long)t;   // float4 index of this thread's element in row r0
    long r = r0;

    // 4-row unroll: 4 independent 128-bit NT loads in flight per thread.
    for (; r + 4 <= r1; r += 4, i += 4L * DVC) {
        v4f a0 = __builtin_nontemporal_load(xv + i);
        v4f a1 = __builtin_nontemporal_load(xv + i + 1L * DVC);
        v4f a2 = __builtin_nontemporal_load(xv + i + 2L * DVC);
        v4f a3 = __builtin_nontemporal_load(xv + i + 3L * DVC);
        __builtin_nontemporal_store(a0 * m, ov + i);
        __builtin_nontemporal_store(a1 * m, ov + i + 1L * DVC);
        __builtin_nontemporal_store(a2 * m, ov + i + 2L * DVC);
        __builtin_nontemporal_store(a3 * m, ov + i + 3L * DVC);
    }
    // Remainder rows
    for (; r < r1; ++r, i += (long)DVC) {
        v4f a = __builtin_nontemporal_load(xv + i);
        __builtin_nontemporal_store(a * m, ov + i);
    }
}

extern "C" void kernel_launch(void* const* d_in, const int* in_sizes, int n_in,
                              void* d_out, int out_size, void* d_ws, size_t ws_size,
                              hipStream_t stream) {
    (void)n_in; (void)d_ws; (void)ws_size; (void)out_size;

    const float* x    = (const float*)d_in[0];  // [B, T, D] f32
    const float* mask = (const float*)d_in[1];  // [D] f32
    float* out        = (float*)d_out;          // [B, T, D] f32

    const long n    = (long)in_sizes[0];        // B*T*D
    const long D    = (long)in_sizes[1];        // 2048
    const long rows = n / D;                    // B*T = 32768

    // 2048 blocks x 16 waves: plenty of parallelism to saturate 23.3 TB/s;
    // each block streams a contiguous strip of rows (16 rows = 256 KB in+out).
    int grid = 2048;
    if ((long)grid > rows) grid = (int)rows;
    int rows_per_block = (int)((rows + grid - 1) / grid);

    dropout_mask_mul_kernel<<<grid, BLOCK_THREADS, 0, stream>>>(
        x, mask, out, rows, rows_per_block);
}